// DeformConv2d_72765335929324
// MI455X (gfx1250) — compile-verified
//
#include <hip/hip_runtime.h>

typedef __attribute__((ext_vector_type(2))) float v2f;
typedef __attribute__((ext_vector_type(8))) float v8f;

#define BATCH  16
#define CIN    64
#define HH     256
#define WW     256
#define COUT   18
#define KS     3
#define TILE_X 64
#define CHPAD  68                 // 64 channels + 4 pad floats (bank-conflict dodge)
#define KTOT   (CIN * KS * KS)    // 576
#define KSTEPS (KTOT / 4)         // 144 wmma k-steps
#define NTILES 2                  // COUT=18 padded to 32 = 2 x 16

// ---- CDNA5 async global->LDS copy (ASYNCcnt-tracked), builtin or asm ------
#if defined(__AMDGCN__) && __has_builtin(__builtin_amdgcn_global_load_async_to_lds_b32)
#define ASYNC_COPY_B32(gp, lp)                                                  \
    __builtin_amdgcn_global_load_async_to_lds_b32(                              \
        (__attribute__((address_space(1))) int*)(gp),                           \
        (__attribute__((address_space(3))) int*)(lp), 0, 0)
#else
#define ASYNC_COPY_B32(gp, lp)                                                  \
    asm volatile("global_load_async_to_lds_b32 %0, %1, off"                     \
                 :: "v"((unsigned)(unsigned long long)(lp)),                    \
                    "v"((unsigned long long)(gp))                               \
                 : "memory")
#endif

#if defined(__AMDGCN__) && __has_builtin(__builtin_amdgcn_s_wait_asynccnt)
#define WAIT_ASYNC() __builtin_amdgcn_s_wait_asynccnt(0)
#else
#define WAIT_ASYNC() asm volatile("s_wait_asynccnt 0x0" ::: "memory")
#endif

// ---------------------------------------------------------------------------
// Pre-pack weights into the exact WMMA B-fragment lane order:
// wpack v2f index = (s*NTILES + nt)*32 + lane, holding
//   {W[n][k0+2*hi], W[n][k0+2*hi+1]}  with n = nt*16 + (lane&15), k0 = 4*s,
//   hi = lane>>4, k = tap*64 + c, weight layout [COUT][CIN][3][3].
// Channels n >= 18 are zero-padded so the second N-tile holds ch 16-17 plus 14 zero columns.
// ---------------------------------------------------------------------------
__global__ void prepack_weights_kernel(const float* __restrict__ weight,
                                       float* __restrict__ wpack) {
    int i = blockIdx.x * blockDim.x + threadIdx.x;
    if (i >= KSTEPS * NTILES * 64) return;
    int s    = i >> 7;        // / (NTILES*64)
    int rem  = i & 127;
    int nt   = rem >> 6;
    int e    = rem & 63;
    int lane = e >> 1;
    int r    = e & 1;
    int hi   = lane >> 4;
    int lo   = lane & 15;
    int n    = nt * 16 + lo;
    int k    = 4 * s + 2 * hi + r;
    int c    = k & (CIN - 1);
    int tap  = k >> 6;
    float v  = 0.0f;
    if (n < COUT) v = weight[(n * CIN + c) * (KS * KS) + tap];
    wpack[i] = v;
}

// ---------------------------------------------------------------------------
// Implicit-GEMM conv: one workgroup per (b, y, 64-px x tile).
// 8 waves: wave = (nt, mt); each wave computes a 16x16 f32 C tile via
// 144 x V_WMMA_F32_16X16X4_F32. Input slab staged via async global->LDS DMA.
// ---------------------------------------------------------------------------
__global__ __launch_bounds__(256)
void deform_conv_wmma_kernel(const float* __restrict__ input,
                             const float* __restrict__ offs,
                             const float* __restrict__ bias,
                             const float* __restrict__ wpack,
                             float* __restrict__ out) {
    __shared__ float lds_in[3 * 66 * CHPAD];   // [row][col][ch] ch-innermost
    __shared__ float lds_c[32 * 66];           // [ch][px] staged C

    const int bidx = blockIdx.x;               // ((b*H + y)*4 + xt)
    const int xt = bidx & 3;
    const int y  = (bidx >> 2) & (HH - 1);
    const int b  = bidx >> 10;
    const int x0 = xt * TILE_X;
    const int tid = threadIdx.x;

    // ---- Stage input slab: rows y-1..y+1, cols x0-1..x0+64, all 64 ch ----
    // Async DMA straight into LDS; per-lane LDS addresses perform the
    // NCHW -> channel-innermost transpose with no VGPR data roundtrip.
    // Zero-pad halo elements with plain LDS stores.
    for (int i = tid; i < 3 * 66 * CIN; i += 256) {
        int col = i % 66;
        int t   = i / 66;
        int row = t % 3;
        int c   = t / 3;
        int yy  = y + row - 1;
        int xx  = x0 + col - 1;
        float* lp = &lds_in[(row * 66 + col) * CHPAD + c];
        if ((unsigned)yy < HH && (unsigned)xx < WW) {
            const float* gp = &input[((b * CIN + c) * HH + yy) * WW + xx];
            ASYNC_COPY_B32(gp, lp);
        } else {
            *lp = 0.0f;
        }
    }
    WAIT_ASYNC();        // drain this wave's ASYNCcnt before workgroup barrier
    __syncthreads();

    const int wave = tid >> 5;
    const int lane = tid & 31;
    const int hi   = lane >> 4;
    const int lo   = lane & 15;
    const int nt   = wave >> 2;            // N tile (0: ch 0-15, 1: ch 16-31 padded)
    const int mt   = wave & 3;             // M tile (16 pixels each)
    const int xlocal = mt * 16 + lo;       // A-matrix row (pixel) for this lane

    const v2f* wp = (const v2f*)wpack;

    v8f acc = {};
    #pragma unroll
    for (int tap = 0; tap < 9; ++tap) {
        const int dy = tap / 3;
        const int dx = tap % 3;
        // A fragment base: pixel (y+dy-1, x0+xlocal+dx-1) -> slab (dy, xlocal+dx)
        const int abase = (dy * 66 + xlocal + dx) * CHPAD + 2 * hi;
        // B fragment base (v2f units): s = tap*16 + cs
        const int wbase = (tap * 16) * (NTILES * 32) + nt * 32 + lane;
        #pragma unroll
        for (int cs = 0; cs < 16; ++cs) {
            v2f a  = *(const v2f*)&lds_in[abase + 4 * cs];   // ds_load (2addr-merged)
            v2f bm = wp[wbase + cs * (NTILES * 32)];         // global_load_b64 (L2-hot)
            acc = __builtin_amdgcn_wmma_f32_16x16x4_f32(
                /*neg_a=*/false, a, /*neg_b=*/false, bm,
                /*c_mod=*/(short)0, acc, /*reuse_a=*/false, /*reuse_b=*/false);
        }
    }

    // ---- C layout: acc[r] is (M = mt*16 + r + 8*hi, N = nt*16 + lo) ----
    const int cm = mt * 16 + 8 * hi;
    const int cn = nt * 16 + lo;
    #pragma unroll
    for (int r = 0; r < 8; ++r)
        lds_c[cn * 66 + cm + r] = acc[r];
    __syncthreads();

    // ---- Epilogue: coalesced stores fused with +offset +bias ----
    for (int i = tid; i < COUT * TILE_X; i += 256) {
        int j    = i >> 6;                  // output channel
        int px   = i & 63;                  // pixel in tile
        int gidx = ((b * COUT + j) * HH + y) * WW + x0 + px;
        out[gidx] = lds_c[j * 66 + px] + offs[gidx] + bias[j];
    }
}

extern "C" void kernel_launch(void* const* d_in, const int* in_sizes, int n_in,
                              void* d_out, int out_size, void* d_ws, size_t ws_size,
                              hipStream_t stream) {
    const float* input  = (const float*)d_in[0];
    const float* offs   = (const float*)d_in[1];
    const float* weight = (const float*)d_in[2];
    const float* bias   = (const float*)d_in[3];
    float* out   = (float*)d_out;
    float* wpack = (float*)d_ws;            // 144*2*64*4 = 73,728 bytes

    const int npack = KSTEPS * NTILES * 64;
    prepack_weights_kernel<<<(npack + 255) / 256, 256, 0, stream>>>(weight, wpack);

    dim3 grid(BATCH * HH * (WW / TILE_X)); // 16384 workgroups
    deform_conv_wmma_kernel<<<grid, 256, 0, stream>>>(input, offs, bias, wpack, out);
}